// MeanAggregator_88519275970844
// MI455X (gfx1250) — compile-verified
//
#include <hip/hip_runtime.h>

#define F      128
#define WAVE   32
#define NSCAN  1024
#define DEPTH  4
#define WPB    8        // waves per block (256 threads)

#if __has_builtin(__builtin_amdgcn_global_load_async_to_lds_b128)
#define USE_ASYNC 1
typedef int v4i __attribute__((ext_vector_type(4)));
typedef __attribute__((address_space(1))) v4i* gas_v4i_p;   // global (as1)
typedef __attribute__((address_space(3))) v4i* las_v4i_p;   // LDS (as3)
#else
#define USE_ASYNC 0
#endif

// ---------------- CSR build ----------------

__global__ void mg_zero(int* p, int n) {
    int i = blockIdx.x * blockDim.x + threadIdx.x;
    if (i < n) p[i] = 0;
}

__global__ void mg_count(const int* __restrict__ dst, int E, int* __restrict__ counts) {
    int e = blockIdx.x * blockDim.x + threadIdx.x;
    if (e < E) atomicAdd(&counts[dst[e]], 1);
}

__global__ void mg_chunksum(const int* __restrict__ counts, int N, int chunk,
                            int* __restrict__ sums) {
    int t = blockIdx.x * blockDim.x + threadIdx.x;
    if (t >= NSCAN) return;
    int b = t * chunk, s = 0;
    for (int i = b; i < b + chunk && i < N; ++i) s += counts[i];
    sums[t] = s;
}

__global__ void mg_scan1(int* sums) {
    if (blockIdx.x == 0 && threadIdx.x == 0) {
        int run = 0;
        for (int t = 0; t < NSCAN; ++t) { int v = sums[t]; sums[t] = run; run += v; }
    }
}

__global__ void mg_offsets(const int* __restrict__ counts, int N, int chunk,
                           const int* __restrict__ sums, int* __restrict__ offs,
                           int* __restrict__ cursor, int E) {
    int t = blockIdx.x * blockDim.x + threadIdx.x;
    if (t >= NSCAN) return;
    int b = t * chunk, run = sums[t];
    for (int i = b; i < b + chunk && i < N; ++i) {
        offs[i] = run; cursor[i] = run; run += counts[i];
    }
    if (t == NSCAN - 1) offs[N] = E;
}

__global__ void mg_fill(const int* __restrict__ dst, const int* __restrict__ src, int E,
                        int* __restrict__ cursor, int* __restrict__ nbr) {
    int e = blockIdx.x * blockDim.x + threadIdx.x;
    if (e < E) {
        int p = atomicAdd(&cursor[dst[e]], 1);
        nbr[p] = src[e];
    }
}

// ---------------- gather-mean: one wave per node ----------------

__global__ void __launch_bounds__(256)
mg_agg(const float* __restrict__ x, const int* __restrict__ offs,
       const int* __restrict__ nbr, float* __restrict__ out, int N) {
#if USE_ASYNC
    __shared__ float stage[WPB][DEPTH][F];   // 16 KB
#endif
    int wid  = threadIdx.x / WAVE;
    int lane = threadIdx.x % WAVE;
    int v = blockIdx.x * WPB + wid;
    if (v >= N) return;

    int b = offs[v], e = offs[v + 1];
    int deg = e - b;
    float4 acc = make_float4(0.f, 0.f, 0.f, 0.f);
    const float* xl = x + lane * 4;

#if USE_ASYNC
    for (int t = b; t < e; t += DEPTH) {
        int m = min(DEPTH, e - t);
        for (int j = 0; j < m; ++j) {
            int u = nbr[t + j];
            float* gsrc = const_cast<float*>(xl + (long)u * F);
            __builtin_amdgcn_global_load_async_to_lds_b128(
                (gas_v4i_p)gsrc,
                (las_v4i_p)&stage[wid][j][lane * 4],
                0, 0);
        }
#if __has_builtin(__builtin_amdgcn_s_wait_asynccnt)
        __builtin_amdgcn_s_wait_asynccnt(0);
        asm volatile("" ::: "memory");
#else
        asm volatile("s_wait_asynccnt 0" ::: "memory");
#endif
        for (int j = 0; j < m; ++j) {
            float4 r = *(const float4*)&stage[wid][j][lane * 4];
            acc.x += r.x; acc.y += r.y; acc.z += r.z; acc.w += r.w;
        }
    }
#else
    int i = b;
    for (; i + 1 < e; i += 2) {
        int u0 = nbr[i], u1 = nbr[i + 1];
        float4 r0 = *(const float4*)(xl + (long)u0 * F);
        float4 r1 = *(const float4*)(xl + (long)u1 * F);
        acc.x += r0.x + r1.x; acc.y += r0.y + r1.y;
        acc.z += r0.z + r1.z; acc.w += r0.w + r1.w;
    }
    if (i < e) {
        int u = nbr[i];
        float4 r = *(const float4*)(xl + (long)u * F);
        acc.x += r.x; acc.y += r.y; acc.z += r.z; acc.w += r.w;
    }
#endif

    float inv = 1.0f / (float)max(deg, 1);
    float4 o = make_float4(acc.x * inv, acc.y * inv, acc.z * inv, acc.w * inv);
    *(float4*)&out[(long)v * F + lane * 4] = o;
}

// ---------------- small-workspace fallback: atomic scatter ----------------

__global__ void mg_scatter(const float* __restrict__ x, const int* __restrict__ src,
                           const int* __restrict__ dst, int E,
                           float* __restrict__ out, int* __restrict__ counts) {
    int w = (blockIdx.x * blockDim.x + threadIdx.x) / WAVE;   // one wave per edge
    int lane = threadIdx.x % WAVE;
    if (w >= E) return;
    int u = src[w], v = dst[w];
    float4 r = *(const float4*)(x + (long)u * F + lane * 4);
    float* o = out + (long)v * F + lane * 4;
    atomicAdd(o + 0, r.x); atomicAdd(o + 1, r.y);
    atomicAdd(o + 2, r.z); atomicAdd(o + 3, r.w);
    if (lane == 0) atomicAdd(&counts[v], 1);
}

__global__ void mg_norm(float* __restrict__ out, const int* __restrict__ counts, int N) {
    int w = (blockIdx.x * blockDim.x + threadIdx.x) / WAVE;
    int lane = threadIdx.x % WAVE;
    if (w >= N) return;
    float inv = 1.0f / (float)max(counts[w], 1);
    float4* p = (float4*)(out + (long)w * F + lane * 4);
    float4 r = *p;
    r.x *= inv; r.y *= inv; r.z *= inv; r.w *= inv;
    *p = r;
}

// ---------------- launcher ----------------

extern "C" void kernel_launch(void* const* d_in, const int* in_sizes, int n_in,
                              void* d_out, int out_size, void* d_ws, size_t ws_size,
                              hipStream_t stream) {
    const float* x   = (const float*)d_in[0];
    const int* esrc  = (const int*)d_in[1];   // JAX x64 disabled -> int32
    const int* edst  = (const int*)d_in[2];
    float* out = (float*)d_out;

    int N = in_sizes[0] / F;
    int E = in_sizes[1];
    int chunk = (N + NSCAN - 1) / NSCAN;

    int NA = (N + 256) & ~255;   // padded, covers N+1
    int EA = (E + 255) & ~255;
    size_t need = (size_t)(3 * (size_t)NA + EA + NSCAN) * sizeof(int);

    int* wsi = (int*)d_ws;

    if (ws_size >= need) {
        int* counts = wsi;
        int* offs   = wsi + NA;
        int* cursor = wsi + 2 * (size_t)NA;
        int* nbr    = wsi + 3 * (size_t)NA;
        int* sums   = wsi + 3 * (size_t)NA + EA;

        mg_zero    <<<(N + 255) / 256, 256, 0, stream>>>(counts, N);
        mg_count   <<<(E + 255) / 256, 256, 0, stream>>>(edst, E, counts);
        mg_chunksum<<<NSCAN / 256, 256, 0, stream>>>(counts, N, chunk, sums);
        mg_scan1   <<<1, 32, 0, stream>>>(sums);
        mg_offsets <<<NSCAN / 256, 256, 0, stream>>>(counts, N, chunk, sums, offs, cursor, E);
        mg_fill    <<<(E + 255) / 256, 256, 0, stream>>>(edst, esrc, E, cursor, nbr);
        mg_agg     <<<(N + WPB - 1) / WPB, 256, 0, stream>>>(x, offs, nbr, out, N);
    } else {
        // atomic fallback: needs only N ints of scratch
        int* counts = wsi;
        int nf = N * F;
        mg_zero   <<<(nf + 255) / 256, 256, 0, stream>>>((int*)out, nf);
        mg_zero   <<<(N + 255) / 256, 256, 0, stream>>>(counts, N);
        mg_scatter<<<((long)E * WAVE + 255) / 256, 256, 0, stream>>>(x, esrc, edst, E, out, counts);
        mg_norm   <<<((long)N * WAVE + 255) / 256, 256, 0, stream>>>(out, counts, N);
    }
}